// CriticalNodeDelayCalculator_37434934952354
// MI455X (gfx1250) — compile-verified
//
#include <hip/hip_runtime.h>
#include <math.h>

#define B_   32
#define C_   32
#define N_   500
#define T_   12
#define KSEL 10
#define NT   (N_*T_)      // 6000 = data's (n,t) flat index, matches GEMM column index

typedef __attribute__((ext_vector_type(2))) float v2f;
typedef __attribute__((ext_vector_type(8))) float v8f;

#ifndef __has_builtin
#define __has_builtin(x) 0
#endif
#if __has_builtin(__builtin_amdgcn_wmma_f32_16x16x4_f32)
#define HAVE_WMMA_F32 1
#else
#define HAVE_WMMA_F32 0
#endif

// ---------------------------------------------------------------------------
// Kernel 1: influence[b,t,n] = sum_m renorm(softmax_m(alpha*x_n*x_m + beta*x_n
//           + gamma*x_m + delta))  (rank-1 logits; faithful 3-pass softmax)
// one wave per row n; x row cached in LDS
// ---------------------------------------------------------------------------
__global__ __launch_bounds__(256) void infl_kernel(
    const float* __restrict__ data,
    const float* __restrict__ fc1_w, const float* __restrict__ fc1_b,
    const float* __restrict__ fc2_w, const float* __restrict__ fc2_b,
    float* __restrict__ infl)
{
  const int bt = blockIdx.x;            // 0..383
  const int b = bt / T_, t = bt % T_;
  __shared__ float xs[N_];
  for (int n = threadIdx.x; n < N_; n += blockDim.x)
    xs[n] = data[((size_t)b*C_*N_ + n)*T_ + t];          // x[b,t,n]=data[b,0,n,t]

  float alpha = 0.f, beta = 0.f, gamma = 0.f, delta = 0.f;
  for (int d = 0; d < 32; ++d) {
    float w1 = fc1_w[d], b1 = fc1_b[d], w2 = fc2_w[d], b2 = fc2_b[d];
    alpha += w1*w2; beta += w1*b2; gamma += b1*w2; delta += b1*b2;
  }
  __syncthreads();

  const int lane  = threadIdx.x & 31;
  const int wave  = threadIdx.x >> 5;
  const int nwav  = blockDim.x >> 5;
  for (int n = wave; n < N_; n += nwav) {
    const float xn  = xs[n];
    const float a   = alpha*xn + gamma;
    const float cst = beta*xn + delta;
    // pass 1: max
    float mx = -INFINITY;
    #pragma unroll
    for (int i = 0; i < 16; ++i) {
      int m = lane + 32*i;
      float v = (m < N_) ? (a*xs[m] + cst) : -INFINITY;
      mx = fmaxf(mx, v);
    }
    for (int off = 16; off; off >>= 1) mx = fmaxf(mx, __shfl_xor(mx, off, 32));
    // pass 2: exp + S1
    float E[16]; float s1 = 0.f;
    #pragma unroll
    for (int i = 0; i < 16; ++i) {
      int m = lane + 32*i;
      float e = (m < N_) ? expf(a*xs[m] + cst - mx) : 0.f;
      E[i] = e; s1 += e;
    }
    for (int off = 16; off; off >>= 1) s1 += __shfl_xor(s1, off, 32);
    // pass 3: S2 = sum(p), influence = sum(p/S2)
    float s2 = 0.f;
    #pragma unroll
    for (int i = 0; i < 16; ++i) s2 += E[i] / s1;
    for (int off = 16; off; off >>= 1) s2 += __shfl_xor(s2, off, 32);
    float fl = 0.f;
    #pragma unroll
    for (int i = 0; i < 16; ++i) fl += (E[i] / s1) / s2;
    for (int off = 16; off; off >>= 1) fl += __shfl_xor(fl, off, 32);
    if (lane == 0) infl[bt*N_ + n] = fl;
  }
}

// ---------------------------------------------------------------------------
// Kernel 2: top-K (ties -> lowest index, descending order) then
//           s[b,t] = sum_k x[b,t,top_idx_k] * lin_w[k]
// ---------------------------------------------------------------------------
__global__ __launch_bounds__(64) void topk_kernel(
    const float* __restrict__ data, const float* __restrict__ infl,
    const float* __restrict__ lin_w, float* __restrict__ s_out)
{
  const int bt = blockIdx.x; const int b = bt / T_, t = bt % T_;
  __shared__ float vals[N_];
  for (int n = threadIdx.x; n < N_; n += blockDim.x) vals[n] = infl[bt*N_ + n];
  __syncthreads();
  if (threadIdx.x == 0) {
    float s = 0.f;
    for (int k = 0; k < KSEL; ++k) {
      float best = -INFINITY; int bi = 0;
      for (int n = 0; n < N_; ++n) { float v = vals[n]; if (v > best) { best = v; bi = n; } }
      vals[bi] = -INFINITY;
      float selv = data[((size_t)b*C_*N_ + bi)*T_ + t];
      s += selv * lin_w[k];
    }
    s_out[bt] = s;
  }
}

// ---------------------------------------------------------------------------
// Kernel 3: tau[b,t,n] = lagc[argmax_w (s[b,t+w]*x[b,t+w,n] + lin_b)]
//           lagc[i] = i + 1 - ceil((T-t)/2)
// ---------------------------------------------------------------------------
__global__ __launch_bounds__(256) void tau_kernel(
    const float* __restrict__ data, const float* __restrict__ s_in,
    const float* __restrict__ lin_b, int* __restrict__ tau)
{
  const int tid = blockIdx.x*blockDim.x + threadIdx.x;
  if (tid >= B_*N_) return;
  const int b = tid / N_, n = tid % N_;
  const float lb = lin_b[0];
  float xr[T_], sb[T_];
  #pragma unroll
  for (int t = 0; t < T_; ++t) {
    xr[t] = data[((size_t)b*C_*N_ + n)*T_ + t];
    sb[t] = s_in[b*T_ + t];
  }
  #pragma unroll
  for (int t = 0; t < T_; ++t) {
    const int win = T_ - t;
    float best = -INFINITY; int bi = 0;
    for (int w = 0; w < win; ++w) {
      float v = sb[t+w]*xr[t+w] + lb;
      if (v > best) { best = v; bi = w; }
    }
    tau[(b*T_ + t)*N_ + n] = bi + 1 - (win + 1)/2;
  }
}

// ---------------------------------------------------------------------------
// Kernel 4: scaled[b,w,n,c] = sum_{t<=w} data[b,c,n, t+((w-t-tau_t) mod T-t)]/(t+1)
//           stored transposed as Dmat[b][c][n*T+w] (top half of GEMM B operand)
// ---------------------------------------------------------------------------
__global__ __launch_bounds__(256) void scaled_kernel(
    const float* __restrict__ data, const int* __restrict__ tau,
    float* __restrict__ Dmat)
{
  const int tid = blockIdx.x*blockDim.x + threadIdx.x;    // exactly B_*C_*N_
  if (tid >= B_*C_*N_) return;
  const int n = tid % N_;
  const int c = (tid / N_) % C_;
  const int b = tid / (N_*C_);
  float row[T_]; int tv[T_];
  const float* dp = data + ((size_t)(b*C_ + c)*N_ + n)*T_;
  #pragma unroll
  for (int t = 0; t < T_; ++t) row[t] = dp[t];
  #pragma unroll
  for (int t = 0; t < T_; ++t) tv[t] = tau[(b*T_ + t)*N_ + n];
  float* op = Dmat + ((size_t)(b*C_ + c)*N_ + n)*T_;
  #pragma unroll
  for (int w = 0; w < T_; ++w) {
    float acc = 0.f;
    for (int t = 0; t <= w; ++t) {
      const int win = T_ - t;
      int r = (w - t - tv[t]) % win; if (r < 0) r += win;
      acc += row[t + r] / (float)(t + 1);
    }
    op[w] = acc;
  }
}

// ---------------------------------------------------------------------------
// Kernel 5: Out[b](32 x 6000) = out_w(32x64) x D[b](64x6000) + out_b
//   D top 32 rows = Dmat[b], bottom 32 rows = data[b] (layout coincides!)
//   One wave computes a 16x16 tile via 16x V_WMMA_F32_16X16X4_F32 (fp32 path).
// ---------------------------------------------------------------------------
__global__ __launch_bounds__(128) void gemm_kernel(
    const float* __restrict__ data, const float* __restrict__ Dmat,
    const float* __restrict__ out_w, const float* __restrict__ out_b,
    float* __restrict__ out)
{
  const int gw   = blockIdx.x*(blockDim.x >> 5) + (threadIdx.x >> 5); // 0..23999
  const int lane = threadIdx.x & 31;
  const int mt   = gw & 1;                  // 2 M-tiles of 16 rows
  const int jt   = (gw >> 1) % (NT/16);     // 375 J-tiles
  const int b    = gw / (2*(NT/16));        // 32 batches
  const int half = lane >> 4, lm = lane & 15;
  const int m0   = mt*16;
  const int j    = jt*16 + lm;

  const float* wrow = out_w + (size_t)(m0 + lm)*64;        // A rows (M=lm)
  const float* Btop = Dmat + (size_t)b*C_*NT;              // c in [0,32)
  const float* Bbot = data + (size_t)b*C_*NT;              // c in [32,64)

  float res[8];
#if HAVE_WMMA_F32
  v8f acc = {};
  // A 16x4 layout: lanes0-15 M=lane, v0:K=0 v1:K=1; lanes16-31: K=2/3.
  // B 4x16 layout: v0: K=0(l0-15)/K=2(l16-31), v1: K=1/K=3; N=lane%16.
  #pragma unroll
  for (int ks = 0; ks < 8; ++ks) {          // K = 0..31 : top half (Dmat)
    const int kb = ks*4 + 2*half;
    v2f A, Bv;
    A.x = wrow[kb];  A.y = wrow[kb + 1];
    const float* src = Btop + (size_t)kb*NT + j;
    Bv.x = src[0];   Bv.y = src[NT];
    acc = __builtin_amdgcn_wmma_f32_16x16x4_f32(false, A, false, Bv,
                                                (short)0, acc, false, false);
  }
  #pragma unroll
  for (int ks = 8; ks < 16; ++ks) {         // K = 32..63 : bottom half (data)
    const int kb = ks*4 + 2*half;
    v2f A, Bv;
    A.x = wrow[kb];  A.y = wrow[kb + 1];
    const float* src = Bbot + (size_t)(kb - 32)*NT + j;
    Bv.x = src[0];   Bv.y = src[NT];
    acc = __builtin_amdgcn_wmma_f32_16x16x4_f32(false, A, false, Bv,
                                                (short)0, acc, false, false);
  }
  #pragma unroll
  for (int r = 0; r < 8; ++r) res[r] = acc[r];
#else
  // VALU fallback (keeps compile green if the f32 WMMA builtin is absent)
  #pragma unroll
  for (int r = 0; r < 8; ++r) {
    const int o = m0 + r + 8*half;
    float a = 0.f;
    for (int c = 0; c < 32; ++c) a += out_w[(size_t)o*64 + c]      * Btop[(size_t)c*NT + j];
    for (int c = 0; c < 32; ++c) a += out_w[(size_t)o*64 + 32 + c] * Bbot[(size_t)c*NT + j];
    res[r] = a;
  }
#endif
  // C/D layout: VGPR r, lanes0-15 -> M=r, lanes16-31 -> M=r+8, N=lane%16
  float* op = out + (size_t)b*C_*NT + (size_t)(m0 + 8*half)*NT + j;
  #pragma unroll
  for (int r = 0; r < 8; ++r) {
    const int o = m0 + r + 8*half;
    op[(size_t)r*NT] = res[r] + out_b[o];
  }
}

// ---------------------------------------------------------------------------
extern "C" void kernel_launch(void* const* d_in, const int* in_sizes, int n_in,
                              void* d_out, int out_size, void* d_ws, size_t ws_size,
                              hipStream_t stream) {
  (void)in_sizes; (void)n_in; (void)out_size; (void)ws_size;
  const float* data  = (const float*)d_in[0];
  const float* fc1_w = (const float*)d_in[1];
  const float* fc1_b = (const float*)d_in[2];
  const float* fc2_w = (const float*)d_in[3];
  const float* fc2_b = (const float*)d_in[4];
  const float* lin_w = (const float*)d_in[5];
  const float* lin_b = (const float*)d_in[6];
  const float* out_w = (const float*)d_in[7];
  const float* out_b = (const float*)d_in[8];
  float* out = (float*)d_out;

  char* ws = (char*)d_ws;
  float* Dmat = (float*)(ws);                                  // 24,576,000 B
  float* infl = (float*)(ws + 24576000);                       //    768,000 B
  float* s    = (float*)(ws + 24576000 + 768000);              //      1,536 B
  int*   tau  = (int*)  (ws + 24576000 + 768000 + 2048);       //    768,000 B

  infl_kernel  <<<B_*T_,                384, 0, stream>>>(data, fc1_w, fc1_b, fc2_w, fc2_b, infl);
  topk_kernel  <<<B_*T_,                 64, 0, stream>>>(data, infl, lin_w, s);
  tau_kernel   <<<(B_*N_ + 255)/256,    256, 0, stream>>>(data, s, lin_b, tau);
  scaled_kernel<<<(B_*C_*N_ + 255)/256, 256, 0, stream>>>(data, tau, Dmat);
  gemm_kernel  <<<(B_*2*(NT/16))/4,     128, 0, stream>>>(data, Dmat, out_w, out_b, out);
}